// SingleHead_75204877353027
// MI455X (gfx1250) — compile-verified
//
#include <hip/hip_runtime.h>

// MI455X / gfx1250, wave32. Matrix math via V_WMMA_F32_16X16X4_F32:
//   A: 16x4 f32  -> v2f per lane (lane&15 = M row, lane>>4 selects K pair)
//   B: 4x16 f32  -> v2f per lane (lane&15 = N col, lane>>4 selects K pair)
//   C/D: 16x16 f32 -> v8f per lane (lane&15 = N, vgpr i = M row, +8 for lanes>=16)

typedef float v2f __attribute__((ext_vector_type(2)));
typedef float v8f __attribute__((ext_vector_type(8)));

static __device__ __forceinline__ v8f wmma_f32(v2f a, v2f b, v8f c) {
  return __builtin_amdgcn_wmma_f32_16x16x4_f32(false, a, false, b, (short)0, c,
                                               false, false);
}

#define Tdim 1024
#define Ddim 1024
#define HSd  64
#define SCHUNK 8  // s-tiles per workgroup chunk (128 s values)
#define NEG_INF (-__builtin_inff())

// Async copy of one 16-byte chunk global -> LDS (ASYNCcnt-tracked).
static __device__ __forceinline__ void async_g2l_b128(unsigned lds_off,
                                                      const float* g) {
  asm volatile("global_load_async_to_lds_b128 %0, %1, off" ::"v"(lds_off),
               "v"(g)
               : "memory");
}
static __device__ __forceinline__ void wait_async_le(int) {}
static __device__ __forceinline__ void wait_async_16() {
  asm volatile("s_wait_asynccnt 0x10" ::: "memory");
}
static __device__ __forceinline__ void wait_async_0() {
  asm volatile("s_wait_asynccnt 0x0" ::: "memory");
}

// ---------------------------------------------------------------------------
// Kernel 1: q/k/v = x @ W + b.  GEMM M=B*T=8192, N=64, K=1024.
// ---------------------------------------------------------------------------
__global__ void __launch_bounds__(384)
proj_qkv_kernel(const float* __restrict__ x,
                const float* __restrict__ Wq, const float* __restrict__ bq,
                const float* __restrict__ Wk, const float* __restrict__ bk,
                const float* __restrict__ Wv, const float* __restrict__ bv,
                float* __restrict__ q, float* __restrict__ k,
                float* __restrict__ v) {
  const int lane = threadIdx.x & 31;
  const int wave = threadIdx.x >> 5;
  const int proj = wave >> 2;        // 0=q, 1=k, 2=v
  const int n0   = (wave & 3) * 16;  // column tile within HS=64
  const int m0   = blockIdx.x * 16;  // row tile within B*T=8192
  const int lm   = lane & 15;
  const int hi   = lane >> 4;

  const float* W    = (proj == 0) ? Wq : (proj == 1) ? Wk : Wv;
  const float* bias = (proj == 0) ? bq : (proj == 1) ? bk : bv;
  float*       dst  = (proj == 0) ? q  : (proj == 1) ? k  : v;

  v8f acc = {};
  const float* xrow = x + (size_t)(m0 + lm) * Ddim + hi * 2;
  const float* wcol = W + (size_t)(hi * 2) * HSd + (n0 + lm);
  for (int kc = 0; kc < Ddim; kc += 4) {
    v2f a = *(const v2f*)(xrow + kc);
    v2f b;
    b.x = wcol[(size_t)kc * HSd];
    b.y = wcol[(size_t)kc * HSd + HSd];
    acc = wmma_f32(a, b, acc);
  }
  const float bn = bias[n0 + lm];
  float* o = dst + (size_t)(m0 + 8 * hi) * HSd + (n0 + lm);
#pragma unroll
  for (int i = 0; i < 8; ++i) o[(size_t)i * HSd] = acc[i] + bn;
}

// ---------------------------------------------------------------------------
// Kernel 2: fused causal attention + relative-position bias, flash-decoding.
// Grid (t_tile, s_chunk); 8 waves = 8 batches so each rel_pos element is
// streamed from HBM exactly once, amortized over all batches. rel tiles are
// double-buffered in dynamic LDS via global_load_async_to_lds_b128, with an
// additive (j+s)&15 rotate chunk swizzle so bias A-fragment ds reads are bank-conflict-free.
// Partial (m, l, unnormalized O) per chunk go to workspace; merged in k3.
// ---------------------------------------------------------------------------

// Issue one 16t-row... (2 rows owned by this wave) rel tile into LDS buffer.
static __device__ __forceinline__ void issue_rel_tile(const float* __restrict__ rel,
                                                      float* lds_base,  // dyn lds
                                                      int buf, int wave, int lane,
                                                      int t0, int s0) {
#pragma unroll
  for (int tl2 = 0; tl2 < 2; ++tl2) {
    const int tl = 2 * wave + tl2;
    const int tg = t0 + tl;
    const float* gbase = rel + ((size_t)tg * Tdim + s0) * HSd;  // 4KB contiguous
    float* lrow = lds_base + buf * 16384 + tl * 1024;           // 16s x 64c
#pragma unroll
    for (int it = 0; it < 8; ++it) {
      const int c16   = lane + 32 * it;          // chunk id 0..255 (16B each)
      const int s_idx = c16 >> 4;                // s row 0..15
      const int j     = c16 & 15;                // 16B chunk within row
      const int fo    = ((j + s_idx) & 15) * 4;  // swizzled float offset
      const unsigned lds_off =
          (unsigned)(size_t)(lrow + s_idx * 64 + fo);
      async_g2l_b128(lds_off, gbase + s_idx * HSd + j * 4);
    }
  }
}

__global__ void __launch_bounds__(256)
attn_chunk_kernel(const float* __restrict__ q, const float* __restrict__ k,
                  const float* __restrict__ v, const float* __restrict__ rel,
                  float* __restrict__ po, float* __restrict__ pm,
                  float* __restrict__ pl) {
  extern __shared__ float relbuf[];      // [2][16t][16s][64c], swizzled, 128KB
  __shared__ float bias_lds[16][16][9];  // [t_local][s][b]
  __shared__ float att_lds[8][16][18];   // [wave][t][s]

  const int bx = blockIdx.x;             // t tile
  const int cy = blockIdx.y;             // s chunk
  if (cy * SCHUNK > bx) return;          // causal: chunk beyond diagonal
  const int st_lo = cy * SCHUNK;
  const int st_hi = min(st_lo + SCHUNK, bx + 1);

  const int lane = threadIdx.x & 31;
  const int wave = threadIdx.x >> 5;  // wave == batch b
  const int lm   = lane & 15;
  const int hi   = lane >> 4;
  const int t0   = bx * 16;
  const int b    = wave;

  float m_run = NEG_INF;
  float l_run = 0.0f;
  v8f oacc[4] = {{}, {}, {}, {}};

  // prologue: start async stream of the first rel tile
  issue_rel_tile(rel, relbuf, st_lo & 1, wave, lane, t0, st_lo * 16);

  for (int st = st_lo; st < st_hi; ++st) {
    const int s0  = st * 16;
    const int buf = st & 1;
    // prefetch next tile into the other buffer while computing this one
    if (st + 1 < st_hi)
      issue_rel_tile(rel, relbuf, (st + 1) & 1, wave, lane, t0, s0 + 16);

    __syncthreads();  // bias_lds from previous iteration fully consumed
    if (st + 1 < st_hi) wait_async_16();  // current tile landed; next in flight
    else                wait_async_0();

    // ---- (1) bias tiles: D[s, b] = rel[t, s_tile, :] @ q[:, t, :]^T
#pragma unroll
    for (int tl2 = 0; tl2 < 2; ++tl2) {
      const int tl = 2 * wave + tl2;
      const int tg = t0 + tl;
      const float* lrow = relbuf + buf * 16384 + tl * 1024 + lm * 64;
      const float* brow = q + ((size_t)(lm & 7) * Tdim + tg) * HSd + hi * 2;
      v8f bd = {};
#pragma unroll
      for (int kc = 0; kc < HSd; kc += 4) {
        const int fo = (((kc >> 2) + lm) & 15) * 4 + hi * 2;  // de-swizzle
        v2f a  = *(const v2f*)(lrow + fo);
        v2f bb = *(const v2f*)(brow + kc);
        bd = wmma_f32(a, bb, bd);
      }
      if (lm < 8) {
#pragma unroll
        for (int i = 0; i < 8; ++i) bias_lds[tl][i + 8 * hi][lm] = bd[i];
      }
    }
    __syncthreads();

    // ---- (2) transposed score tile ST[s, t] for own batch
    v8f sd = {};
    const float* krow = k + ((size_t)b * Tdim + (s0 + lm)) * HSd + hi * 2;
    const float* qrow = q + ((size_t)b * Tdim + (t0 + lm)) * HSd + hi * 2;
#pragma unroll
    for (int kc = 0; kc < HSd; kc += 4) {
      v2f a  = *(const v2f*)(krow + kc);
      v2f bb = *(const v2f*)(qrow + kc);
      sd = wmma_f32(a, bb, sd);
    }

    // ---- (3) combine, causal mask, online softmax
    const int trow = t0 + lm;
    float p[8];
    float tmax = NEG_INF;
#pragma unroll
    for (int i = 0; i < 8; ++i) {
      const int s = s0 + i + 8 * hi;
      float sc = 8.0f * sd[i] + bias_lds[lm][i + 8 * hi][b];  // sqrt(64)=8
      sc = (s <= trow) ? sc : NEG_INF;
      p[i] = sc;
      tmax = fmaxf(tmax, sc);
    }
    tmax = fmaxf(tmax, __shfl_xor(tmax, 16));
    const float m_new = fmaxf(m_run, tmax);
    const float alpha = __expf(m_run - m_new);
    float psum = 0.0f;
#pragma unroll
    for (int i = 0; i < 8; ++i) {
      p[i] = __expf(p[i] - m_new);
      psum += p[i];
    }
    psum += __shfl_xor(psum, 16);
    l_run = l_run * alpha + psum;
    m_run = m_new;
#pragma unroll
    for (int nt = 0; nt < 4; ++nt)
#pragma unroll
      for (int i = 0; i < 8; ++i) oacc[nt][i] *= alpha;

    // P tile -> LDS as [t][s] to form av A-fragments
#pragma unroll
    for (int i = 0; i < 8; ++i) att_lds[wave][lm][i + 8 * hi] = p[i];
    __builtin_amdgcn_wave_barrier();

    // ---- (4) O[t, h] += P[t, s] @ v[s, h]
#pragma unroll
    for (int kc = 0; kc < 16; kc += 4) {
      v2f a = *(const v2f*)&att_lds[wave][lm][kc + hi * 2];
#pragma unroll
      for (int nt = 0; nt < 4; ++nt) {
        const int h = nt * 16 + lm;
        const float* vp = v + ((size_t)b * Tdim + s0 + kc + hi * 2) * HSd + h;
        v2f bb;
        bb.x = vp[0];
        bb.y = vp[HSd];
        oacc[nt] = wmma_f32(a, bb, oacc[nt]);
      }
    }
  }

  // ---- write partial results (unnormalized O, per-row m and l)
  if (hi == 0) {
    const size_t ml = ((size_t)b * Tdim + t0 + lm) * SCHUNK + cy;
    pm[ml] = m_run;
    pl[ml] = l_run;
  }
#pragma unroll
  for (int nt = 0; nt < 4; ++nt) {
    const int h = nt * 16 + lm;
#pragma unroll
    for (int i = 0; i < 8; ++i) {
      const size_t row = (size_t)b * Tdim + t0 + i + 8 * hi;
      po[(row * SCHUNK + cy) * HSd + h] = oacc[nt][i];
    }
  }
}

// ---------------------------------------------------------------------------
// Kernel 3: merge partial softmax chunks: out = sum_c O_c*e^(m_c-m*) / sum_c
// l_c*e^(m_c-m*).  One thread per output element.
// ---------------------------------------------------------------------------
__global__ void __launch_bounds__(256)
merge_kernel(const float* __restrict__ po, const float* __restrict__ pm,
             const float* __restrict__ pl, float* __restrict__ out) {
  const int idx = blockIdx.x * blockDim.x + threadIdx.x;
  const int h = idx & 63;
  const int t = (idx >> 6) & (Tdim - 1);
  const int b = idx >> 16;
  const int nc = t / (16 * SCHUNK) + 1;  // active chunks for this row
  const size_t ml = ((size_t)b * Tdim + t) * SCHUNK;

  float mstar = NEG_INF;
  for (int c = 0; c < nc; ++c) mstar = fmaxf(mstar, pm[ml + c]);
  float denom = 0.0f, acc = 0.0f;
  for (int c = 0; c < nc; ++c) {
    const float w = __expf(pm[ml + c] - mstar);
    denom += pl[ml + c] * w;
    acc += po[(ml + c) * HSd + h] * w;
  }
  out[idx] = acc / denom;
}

// ---------------------------------------------------------------------------
extern "C" void kernel_launch(void* const* d_in, const int* in_sizes, int n_in,
                              void* d_out, int out_size, void* d_ws,
                              size_t ws_size, hipStream_t stream) {
  const float* x   = (const float*)d_in[0];
  const float* Wq  = (const float*)d_in[1];
  const float* bq  = (const float*)d_in[2];
  const float* Wk  = (const float*)d_in[3];
  const float* bk  = (const float*)d_in[4];
  const float* Wv  = (const float*)d_in[5];
  const float* bv  = (const float*)d_in[6];
  const float* rel = (const float*)d_in[7];
  float* out = (float*)d_out;

  // workspace layout (fp32): q,k,v [8192,64]; po [8,1024,8,64]; pm/pl [8,1024,8]
  float* qw = (float*)d_ws;
  float* kw = qw + (size_t)8192 * HSd;
  float* vw = kw + (size_t)8192 * HSd;
  float* po = vw + (size_t)8192 * HSd;
  float* pm = po + (size_t)8 * Tdim * SCHUNK * HSd;
  float* pl = pm + (size_t)8 * Tdim * SCHUNK;

  proj_qkv_kernel<<<dim3(512), dim3(384), 0, stream>>>(x, Wq, bq, Wk, bk, Wv,
                                                       bv, qw, kw, vw);
  // 128KB dynamic LDS: double-buffered rel tiles [2][16][16][64] f32
  attn_chunk_kernel<<<dim3(64, SCHUNK), dim3(256), 131072, stream>>>(
      qw, kw, vw, rel, po, pm, pl);
  merge_kernel<<<dim3((8 * Tdim * HSd) / 256), dim3(256), 0, stream>>>(po, pm,
                                                                       pl, out);
}